// vector_quantizer_38405597561718
// MI455X (gfx1250) — compile-verified
//
#include <hip/hip_runtime.h>

// Vector quantizer (soft assignment + straight-through), CDNA5 / gfx1250.
//   x: (B,C,H,W) fp32 viewed as (M=524288, 4); center: (256, 4) fp32.
//   out[v] = (center[argmin_k dist2] - w_soft) + w_soft,
//   w_soft = softmax(-dist2) @ center,  dist2 = |x|^2 + |c|^2 - 2 x.c
//
// Per-wave tile: 16 vectors. Distances via V_WMMA_F32_16X16X4_F32 with
//   A = x (16x4, raw), B = (2*log2e)*center^T (4x16), C = -log2e*(|x|^2+|c|^2)
// so D = -log2e * dist2 and p = exp2(D) = exp(-dist2) directly.
// Final per-row reductions use a reduce-scatter butterfly (xor 8,4,2,1) that
// leaves row (l>>1)&7 on lane l of each 16-lane half -> even lanes write.

typedef float v2f __attribute__((ext_vector_type(2)));
typedef float v8f __attribute__((ext_vector_type(8)));

static __device__ __forceinline__ float xor_f(float v, int m) {
  return __shfl_xor(v, m, 32);
}
static __device__ __forceinline__ int xor_i(int v, int m) {
  return __shfl_xor(v, m, 32);
}

// Reduce-scatter sum over 16 lanes of a half: v[8] rows -> lane (l>>1)&7
// holds total of its row in v[0].
static __device__ __forceinline__ void rs_sum(float v[8], bool b3, bool b2, bool b1) {
#pragma unroll
  for (int j = 0; j < 4; ++j) {
    float s = b3 ? v[j] : v[j + 4];
    float r = xor_f(s, 8);
    v[j] = (b3 ? v[j + 4] : v[j]) + r;
  }
#pragma unroll
  for (int j = 0; j < 2; ++j) {
    float s = b2 ? v[j] : v[j + 2];
    float r = xor_f(s, 4);
    v[j] = (b2 ? v[j + 2] : v[j]) + r;
  }
  {
    float s = b1 ? v[0] : v[1];
    float r = xor_f(s, 2);
    v[0] = (b1 ? v[1] : v[0]) + r;
  }
  v[0] += xor_f(v[0], 1);
}

// Reduce-scatter argmax (value fb, index fi; ties -> smaller index).
static __device__ __forceinline__ void rs_amax(float fb[8], int fi[8],
                                               bool b3, bool b2, bool b1) {
#pragma unroll
  for (int j = 0; j < 4; ++j) {
    float sv = b3 ? fb[j] : fb[j + 4];
    int   si = b3 ? fi[j] : fi[j + 4];
    float rv = xor_f(sv, 8);
    int   ri = xor_i(si, 8);
    float kv = b3 ? fb[j + 4] : fb[j];
    int   ki = b3 ? fi[j + 4] : fi[j];
    bool take = (rv > kv) || (rv == kv && ri < ki);
    fb[j] = take ? rv : kv;
    fi[j] = take ? ri : ki;
  }
#pragma unroll
  for (int j = 0; j < 2; ++j) {
    float sv = b2 ? fb[j] : fb[j + 2];
    int   si = b2 ? fi[j] : fi[j + 2];
    float rv = xor_f(sv, 4);
    int   ri = xor_i(si, 4);
    float kv = b2 ? fb[j + 2] : fb[j];
    int   ki = b2 ? fi[j + 2] : fi[j];
    bool take = (rv > kv) || (rv == kv && ri < ki);
    fb[j] = take ? rv : kv;
    fi[j] = take ? ri : ki;
  }
  {
    float sv = b1 ? fb[0] : fb[1];
    int   si = b1 ? fi[0] : fi[1];
    float rv = xor_f(sv, 2);
    int   ri = xor_i(si, 2);
    float kv = b1 ? fb[1] : fb[0];
    int   ki = b1 ? fi[1] : fi[0];
    bool take = (rv > kv) || (rv == kv && ri < ki);
    fb[0] = take ? rv : kv;
    fi[0] = take ? ri : ki;
  }
  {
    float rv = xor_f(fb[0], 1);
    int   ri = xor_i(fi[0], 1);
    bool take = (rv > fb[0]) || (rv == fb[0] && ri < fi[0]);
    fb[0] = take ? rv : fb[0];
    fi[0] = take ? ri : fi[0];
  }
}

__global__ __launch_bounds__(256, 2)
void vq_soft_wmma_kernel(const float* __restrict__ x,
                         const float* __restrict__ center,
                         float* __restrict__ out,
                         int numTiles)   // numVec / 16
{
  constexpr float LOG2E = 1.44269504088896340736f;

  // Codebook staged per block.
  //  lBC[g][lane] = { 2*log2e*c[d0], 2*log2e*c[d1], log2e*|c|^2, 0 }
  //    where (d0,d1) = (0,1) for lanes 0-15 and (2,3) for lanes 16-31
  //    (B-matrix layout mirroring the documented f32 A layout).
  __shared__ float4 lBC[16][32];
  __shared__ float4 lCen[256];      // raw centers (w_soft weights + w_hard gather)

  const int tid = threadIdx.x;      // blockDim.x == 256
  {
    const float4 c = ((const float4*)center)[tid];
    lCen[tid] = c;
    const float n2l = LOG2E * (c.x * c.x + c.y * c.y + c.z * c.z + c.w * c.w);
    const int g  = tid >> 4;
    const int cl = tid & 15;
    const float s = 2.0f * LOG2E;
    lBC[g][cl]      = make_float4(s * c.x, s * c.y, n2l, 0.0f);
    lBC[g][cl + 16] = make_float4(s * c.z, s * c.w, n2l, 0.0f);
  }
  __syncthreads();

  const int lane = tid & 31;
  const int col  = lane & 15;       // D-layout column within a 16-center group
  const int half = lane >> 4;       // 0: rows 0-7, 1: rows 8-15
  const bool b3 = (lane & 8) != 0;
  const bool b2 = (lane & 4) != 0;
  const bool b1 = (lane & 2) != 0;
  const int gw   = blockIdx.x * (blockDim.x >> 5) + (tid >> 5);
  const int nw   = gridDim.x * (blockDim.x >> 5);

  for (int t = gw; t < numTiles; t += nw) {
    const int r0 = t << 4;

    // Prefetch next tile of x for this wave (speculative, OOB-safe).
    __builtin_prefetch(x + 4u * (unsigned)(r0 + (nw << 4)), 0, 1);

    // --- A-matrix load: lane L -> x[r0 + L%16][half?2:0 .. +1]  (b64, coalesced)
    v2f a = *(const v2f*)(x + 4 * (r0 + col) + 2 * half);

    // --- per-row |x|^2 in D layout
    float part = a.x * a.x + a.y * a.y;          // dims {0,1} or {2,3}
    float x2full = part + xor_f(part, 16);       // lane L: |x[r0+L%16]|^2
    float xl[8];
#pragma unroll
    for (int j = 0; j < 8; ++j)
      xl[j] = -LOG2E * __shfl(x2full, j + 8 * half, 32);  // row j + 8*half

    // --- accumulators (per lane: 8 rows of its column residues)
    float den[8], w0[8], w1[8], w2[8], w3[8], fb[8];
    int   fi[8];
#pragma unroll
    for (int j = 0; j < 8; ++j) {
      den[j] = 0.f; w0[j] = 0.f; w1[j] = 0.f; w2[j] = 0.f; w3[j] = 0.f;
      fb[j] = -3.0e38f; fi[j] = 0;
    }

    // --- 16 groups x 16 centers: one WMMA + exp/FMA stream per group
#pragma unroll 2
    for (int g = 0; g < 16; ++g) {
      float4 bc  = lBC[g][lane];                  // ds_read_b128
      float4 cen = lCen[(g << 4) + col];          // ds_read_b128
      v2f b; b.x = bc.x; b.y = bc.y;
      const float c2l = bc.z;

      v8f cin;
#pragma unroll
      for (int j = 0; j < 8; ++j) cin[j] = xl[j] - c2l;

      // D = -log2e * dist2 (16x16 tile)
      v8f dmat = __builtin_amdgcn_wmma_f32_16x16x4_f32(
          false, a, false, b, (short)0, cin, false, false);

      const int kidx = (g << 4) + col;
#pragma unroll
      for (int j = 0; j < 8; ++j) {
        float dv = dmat[j];
        float p  = __builtin_amdgcn_exp2f(dv);    // exp(-dist2)
        den[j] += p;
        w0[j] = fmaf(p, cen.x, w0[j]);
        w1[j] = fmaf(p, cen.y, w1[j]);
        w2[j] = fmaf(p, cen.z, w2[j]);
        w3[j] = fmaf(p, cen.w, w3[j]);
        bool gt = dv > fb[j];                     // argmax D == argmin dist2
        fb[j] = gt ? dv : fb[j];
        fi[j] = gt ? kidx : fi[j];
      }
    }

    // --- reduce-scatter over the 16 columns of each half:
    //     lane l ends holding row (l>>1)&7 (+8 for upper half) in index 0.
    rs_sum(den, b3, b2, b1);
    rs_sum(w0, b3, b2, b1);
    rs_sum(w1, b3, b2, b1);
    rs_sum(w2, b3, b2, b1);
    rs_sum(w3, b3, b2, b1);
    rs_amax(fb, fi, b3, b2, b1);

    // --- 16 writer lanes (even lanes) emit 16 rows (b128 stores)
    if ((lane & 1) == 0) {
      float inv = 1.0f / den[0];
      float sw0 = w0[0] * inv, sw1 = w1[0] * inv;
      float sw2 = w2[0] * inv, sw3 = w3[0] * inv;
      float4 hc = lCen[fi[0]];                    // w_hard gather
      float4 o;
      o.x = (hc.x - sw0) + sw0;                   // straight-through forward value
      o.y = (hc.y - sw1) + sw1;
      o.z = (hc.z - sw2) + sw2;
      o.w = (hc.w - sw3) + sw3;
      const int row = r0 + ((lane & 15) >> 1) + 8 * half;
      ((float4*)out)[row] = o;
    }
  }
}

extern "C" void kernel_launch(void* const* d_in, const int* in_sizes, int n_in,
                              void* d_out, int out_size, void* d_ws, size_t ws_size,
                              hipStream_t stream) {
  const float* x      = (const float*)d_in[0];   // (8,64,64,64) fp32
  const float* center = (const float*)d_in[1];   // (256,4) fp32
  float* out = (float*)d_out;

  const int numVec   = in_sizes[0] / 4;          // 524288
  const int numTiles = numVec / 16;              // 32768 (divisible)

  const int wavesPerBlk = 8;                     // 256 threads
  int blocks = (numTiles + wavesPerBlk - 1) / wavesPerBlk;
  if (blocks > 2048) blocks = 2048;              // ~2 tiles per wave
  if (blocks < 1) blocks = 1;

  vq_soft_wmma_kernel<<<blocks, 256, 0, stream>>>(x, center, out, numTiles);
}